// FeatureAttentionPerFeature_23957327577714
// MI455X (gfx1250) — compile-verified
//
#include <hip/hip_runtime.h>

// ---------------------------------------------------------------------------
// FeatureAttentionPerFeature for MI455X (gfx1250, wave32, WMMA)
//
// scores[b,f] = feat_b . (Wk_f^T q)            (prologue kernel, tiny)
// attn        = softmax_f(scores / sqrt(128))
// combined_b  = sum_f (attn[b,f] * feat_b) @ Wv_f^T   (fp32 WMMA 16x16x4)
// ---------------------------------------------------------------------------

typedef float v2f __attribute__((ext_vector_type(2)));
typedef float v8f __attribute__((ext_vector_type(8)));

#define NFEAT   8
#define HIDDEN  128
#define BATCH   32768

constexpr int kDims[NFEAT] = {32, 64, 96, 128, 192, 256, 384, 512};
constexpr int kOffs[NFEAT] = {0, 32, 96, 192, 320, 512, 768, 1152};
constexpr int TOTD    = 1664;          // sum of feature dims
constexpr int FSTRIDE = TOTD + 4;      // 1668: not a multiple of 64 -> no bank alias
constexpr float kInvSqrtH = 0.0883883476483184406f;   // 1/sqrt(128)

struct PParams {
    const float* wk[NFEAT];
    const float* q;
    float*       p;        // [TOTD] = concat_f (Wk_f^T q)
};

struct AParams {
    const float* feat[NFEAT];
    const float* wv[NFEAT];
    const float* p;
    float*       outC;     // (BATCH, HIDDEN)
    float*       outA;     // (BATCH, NFEAT)
};

// ---------------------------------------------------------------------------
// Prologue: p_f[d] = sum_h Wk_f[h][d] * q[h]   (8 blocks, trivially small)
// ---------------------------------------------------------------------------
__global__ __launch_bounds__(128) void proj_query_kernel(PParams P) {
    const int f  = blockIdx.x;
    const int df = kDims[f];
    const float* __restrict__ W = P.wk[f];
    __shared__ float qs[HIDDEN];
    for (int h = threadIdx.x; h < HIDDEN; h += blockDim.x) qs[h] = P.q[h];
    __syncthreads();
    for (int d = threadIdx.x; d < df; d += blockDim.x) {
        float acc = 0.f;
#pragma unroll 4
        for (int h = 0; h < HIDDEN; ++h) acc = fmaf(W[h * df + d], qs[h], acc);
        P.p[kOffs[f] + d] = acc;
    }
}

// ---------------------------------------------------------------------------
// Main fused kernel: one workgroup = 16 batch rows, 8 waves.
//   wave f : score for feature f
//   wave w : H-column tile [16w, 16w+16) of the combined output
// ---------------------------------------------------------------------------
__global__ __launch_bounds__(256) void feature_attention_kernel(AParams P) {
    extern __shared__ float smem[];
    float* ldsF = smem;                    // [16][FSTRIDE] feature rows
    float* ldsA = smem + 16 * FSTRIDE;     // [16][NFEAT]   scores -> attn

    const int tid  = threadIdx.x;
    const int wave = tid >> 5;             // 0..7
    const int lane = tid & 31;             // wave32
    const int r0   = blockIdx.x * 16;      // batch row base

    // ---- stage 16 x 1664 feature elements into LDS (single global read) ----
#pragma unroll
    for (int f = 0; f < NFEAT; ++f) {
        const int df  = kDims[f];
        const int df4 = df >> 2;
        const float4* __restrict__ src = reinterpret_cast<const float4*>(P.feat[f]);
        for (int idx = tid; idx < 16 * df4; idx += 256) {
            const int row = idx / df4;
            const int c4  = idx - row * df4;
            float4 v = src[(size_t)(r0 + row) * df4 + c4];
            float* dst = &ldsF[row * FSTRIDE + kOffs[f] + c4 * 4];
            dst[0] = v.x; dst[1] = v.y; dst[2] = v.z; dst[3] = v.w;
        }
    }
    __syncthreads();

    // ---- scores: wave f handles feature f; lanes split rows x dim-halves ----
    {
        const int f    = wave;
        const int df   = kDims[f];
        const int off  = kOffs[f];
        const int row  = lane & 15;
        const int half = df >> 1;
        const float* __restrict__ pf = P.p + off + (lane >> 4) * half;
        const float* fr = ldsF + row * FSTRIDE + off + (lane >> 4) * half;
        float acc = 0.f;
#pragma unroll 4
        for (int d = 0; d < half; ++d) acc = fmaf(fr[d], pf[d], acc);
        acc += __shfl_xor(acc, 16, 32);                 // combine the two halves
        if (lane < 16) ldsA[row * NFEAT + f] = acc * kInvSqrtH;
    }
    __syncthreads();

    // ---- softmax over the 8 features (wave 0, one lane per batch row) ----
    if (wave == 0 && lane < 16) {
        float s[NFEAT];
        float m = -1e30f;
#pragma unroll
        for (int f = 0; f < NFEAT; ++f) { s[f] = ldsA[lane * NFEAT + f]; m = fmaxf(m, s[f]); }
        float sum = 0.f;
#pragma unroll
        for (int f = 0; f < NFEAT; ++f) { s[f] = __expf(s[f] - m); sum += s[f]; }
        const float inv = 1.f / sum;
#pragma unroll
        for (int f = 0; f < NFEAT; ++f) {
            const float a = s[f] * inv;
            ldsA[lane * NFEAT + f] = a;
            P.outA[(size_t)(r0 + lane) * NFEAT + f] = a;
        }
    }
    __syncthreads();

    // ---- fold attn weights into the A operand rows (done once, reused 8x) ----
#pragma unroll
    for (int f = 0; f < NFEAT; ++f) {
        const int df4 = kDims[f] >> 2;
        for (int idx = tid; idx < 16 * df4; idx += 256) {
            const int row = idx / df4;
            const int c4  = idx - row * df4;
            const float a = ldsA[row * NFEAT + f];
            float* e = &ldsF[row * FSTRIDE + kOffs[f] + c4 * 4];
            e[0] *= a; e[1] *= a; e[2] *= a; e[3] *= a;
        }
    }
    __syncthreads();

    // ---- combined = sum_f A_f @ Wv_f^T via v_wmma_f32_16x16x4_f32 ----
    // A 16x4 layout: lanes 0-15 rows M, VGPR0={K0,K2},VGPR1={K1,K3} (hi = lane>>4
    // selects K base 0 vs 2).  B 4x16 mirrored: lane%16 = N column of Wv row.
    const int n0 = wave * 16;              // H-column tile
    const int m  = lane & 15;
    const int hi = lane >> 4;              // 0: K{0,1}, 1: K{2,3}
    v8f c = {};                            // 16x16 f32 accumulator (8 VGPRs)
#pragma unroll
    for (int f = 0; f < NFEAT; ++f) {
        const int df = kDims[f];
        const float* __restrict__ arow = ldsF + m * FSTRIDE + kOffs[f] + 2 * hi;
        const float* __restrict__ brow = P.wv[f] + (size_t)(n0 + m) * df + 2 * hi;
#pragma unroll 4
        for (int k = 0; k < df; k += 4) {
            v2f a = *reinterpret_cast<const v2f*>(arow + k);   // ds_load_b64
            v2f b = *reinterpret_cast<const v2f*>(brow + k);   // global b64 (L2-hot)
            c = __builtin_amdgcn_wmma_f32_16x16x4_f32(
                    /*neg_a=*/false, a, /*neg_b=*/false, b,
                    /*c_mod=*/(short)0, c, /*reuse_a=*/false, /*reuse_b=*/false);
        }
    }

    // ---- D layout: VGPR v -> M = v + 8*hi, N = lane%16 ----
#pragma unroll
    for (int v = 0; v < 8; ++v) {
        P.outC[(size_t)(r0 + v + 8 * hi) * HIDDEN + n0 + m] = c[v];
    }
}

// ---------------------------------------------------------------------------
extern "C" void kernel_launch(void* const* d_in, const int* in_sizes, int n_in,
                              void* d_out, int out_size, void* d_ws, size_t ws_size,
                              hipStream_t stream) {
    (void)in_sizes; (void)n_in; (void)out_size; (void)ws_size;

    PParams pp;
    AParams ap;
    for (int i = 0; i < NFEAT; ++i) {
        ap.feat[i] = (const float*)d_in[i];
        pp.wk[i]   = (const float*)d_in[NFEAT + i];
        ap.wv[i]   = (const float*)d_in[2 * NFEAT + i];
    }
    pp.q = (const float*)d_in[3 * NFEAT];

    float* p = (float*)d_ws;               // TOTD floats of scratch
    pp.p = p;
    ap.p = p;
    ap.outC = (float*)d_out;
    ap.outA = (float*)d_out + (size_t)BATCH * HIDDEN;

    proj_query_kernel<<<NFEAT, 128, 0, stream>>>(pp);

    const size_t smemBytes = (size_t)(16 * FSTRIDE + 16 * NFEAT) * sizeof(float);
    feature_attention_kernel<<<BATCH / 16, 256, smemBytes, stream>>>(ap);
}